// Embedder_76261439307872
// MI455X (gfx1250) — compile-verified
//
#include <hip/hip_runtime.h>
#include <hip/hip_bf16.h>

// ---------------------------------------------------------------------------
// RGCN embedder for MI455X (gfx1250, wave32).
// FP32 throughout (problem is memory/atomic bound; ~450MB traffic @23.3TB/s
// dominates the ~5.4 GFLOP of dense math), dense math on V_WMMA_F32_16X16X4_F32.
// ---------------------------------------------------------------------------

typedef float v2f __attribute__((ext_vector_type(2)));
typedef float v8f __attribute__((ext_vector_type(8)));

#define RELS 19
#define DIM  32

// ---------------------------------------------------------------------------
// Kernel 0: zero workspace (cnt + h2acc). Must run every launch: the harness
// poisons ws once and graph-replays, so accumulators need explicit re-init.
// ---------------------------------------------------------------------------
__global__ void k_zero_f32(float* __restrict__ p, size_t n) {
  size_t i = (size_t)blockIdx.x * blockDim.x + threadIdx.x;
  size_t stride = (size_t)gridDim.x * blockDim.x;
  for (; i < n; i += stride) p[i] = 0.0f;
}

// ---------------------------------------------------------------------------
// Kernel 1: layer-1 aggregation. h = ones(N,1), so segment-sum == edge count
// per (relation, dst). One f32 atomic per edge.
// ---------------------------------------------------------------------------
__global__ void k_count_edges(const int* __restrict__ dst,
                              const int* __restrict__ etype,
                              float* __restrict__ cnt, int E, int N) {
  int e = blockIdx.x * blockDim.x + threadIdx.x;
  if (e < E) atomicAdd(&cnt[(size_t)etype[e] * N + dst[e]], 1.0f);
}

// ---------------------------------------------------------------------------
// Kernel 2: layer-1 combine: h1[v,d] = relu(b1[d] + sum_r cnt[r,v]*W1[r,d]).
// One wave per node (lane = feature): cnt[r*N+v] is a uniform (broadcast)
// load per wave; W1 (608 floats) lives in LDS.
// ---------------------------------------------------------------------------
__global__ void k_layer1(const float* __restrict__ cnt,
                         const float* __restrict__ W1,
                         const float* __restrict__ b1,
                         float* __restrict__ h1, int N) {
  __shared__ float Wl[RELS * DIM];
  for (int i = threadIdx.x; i < RELS * DIM; i += blockDim.x) Wl[i] = W1[i];
  __syncthreads();
  const int idx = blockIdx.x * blockDim.x + threadIdx.x;
  const int v = idx >> 5;
  const int d = idx & 31;
  if (v < N) {
    float acc = b1[d];
#pragma unroll
    for (int r = 0; r < RELS; ++r)
      acc = fmaf(cnt[(size_t)r * N + v], Wl[r * DIM + d], acc);
    h1[(size_t)v * DIM + d] = fmaxf(acc, 0.0f);
  }
}

// ---------------------------------------------------------------------------
// Layer-2 tile emit: 16 gathered h1 rows (A, 16x32 f32) x W2[r] (B, 32x32)
// via 16x V_WMMA_F32_16X16X4_F32, then atomic-scatter of the 16x32 D tile
// into h2acc[dst]. Invalid (padding) rows carry zero A data and skip atomics.
// A layout (16x4 f32): lane L -> M=L%16; VGPR0 = K=kb+2*(L/16), VGPR1 = +1.
// D layout (16x16 f32): lane L -> N=L%16; VGPR v -> M = v + 8*(L/16).
// ---------------------------------------------------------------------------
__device__ __forceinline__ void rgcn_emit_tile(
    const float* __restrict__ h1, float* __restrict__ h2acc,
    float* __restrict__ Ah, const int* __restrict__ qsrc,
    const int* __restrict__ qdst, const v2f (*bfrag)[2],
    int valid, int lane) {
  const int l15 = lane & 15;
  const int half = lane >> 4;

  // Stage A rows in LDS (coalesced 128B gathers), zero-pad invalid rows.
#pragma unroll 4
  for (int i = 0; i < 16; ++i) {
    float val = 0.0f;
    if (i < valid) val = h1[(size_t)qsrc[i] * DIM + lane];
    Ah[i * DIM + lane] = val;
  }
  __syncthreads();

  v8f c0 = {};
  v8f c1 = {};
#pragma unroll
  for (int ks = 0; ks < 8; ++ks) {
    const int kb = ks * 4 + 2 * half;
    v2f a;
    a.x = Ah[l15 * DIM + kb];
    a.y = Ah[l15 * DIM + kb + 1];
    c0 = __builtin_amdgcn_wmma_f32_16x16x4_f32(false, a, false, bfrag[ks][0],
                                               (short)0, c0, false, false);
    c1 = __builtin_amdgcn_wmma_f32_16x16x4_f32(false, a, false, bfrag[ks][1],
                                               (short)0, c1, false, false);
  }

#pragma unroll
  for (int v = 0; v < 8; ++v) {
    const int row = v + 8 * half;
    if (row < valid) {
      float* base = h2acc + (size_t)qdst[row] * DIM;
      atomicAdd(base + l15, c0[v]);
      atomicAdd(base + 16 + l15, c1[v]);
    }
  }
  __syncthreads();
}

// ---------------------------------------------------------------------------
// Kernel 3: layer-2 message GEMM. blockIdx.y = relation r; single-wave blocks.
// Ballot-compact edges with etype==r into an LDS queue; every 16 edges, run a
// WMMA tile. B-fragments for W2[r] are hoisted into VGPRs once per block.
// ---------------------------------------------------------------------------
__global__ __launch_bounds__(32) void k_layer2_wmma(
    const int* __restrict__ src, const int* __restrict__ dst,
    const int* __restrict__ etype, const float* __restrict__ h1,
    const float* __restrict__ W2, float* __restrict__ h2acc, int E) {
  const int r = blockIdx.y;
  const int lane = threadIdx.x;
  const int l15 = lane & 15;
  const int half = lane >> 4;

  __shared__ float Wl[DIM * DIM];   // W2[r], 4KB
  __shared__ float Ah[16 * DIM];    // staged A tile, 2KB
  __shared__ int qsrc[64];
  __shared__ int qdst[64];

  for (int i = lane; i < DIM * DIM; i += 32)
    Wl[i] = W2[(size_t)r * DIM * DIM + i];
  __syncthreads();

  // B layout (4x16 f32, KxN): lane L -> N=L%16; VGPR0 = K=kb+2*(L/16), VGPR1 = +1.
  v2f bfrag[8][2];
#pragma unroll
  for (int ks = 0; ks < 8; ++ks) {
    const int kb = ks * 4 + 2 * half;
#pragma unroll
    for (int nt = 0; nt < 2; ++nt) {
      const int n = nt * 16 + l15;
      bfrag[ks][nt].x = Wl[(kb + 0) * DIM + n];
      bfrag[ks][nt].y = Wl[(kb + 1) * DIM + n];
    }
  }

  int qcount = 0;  // wave-uniform (derived from ballot)
  const int stride = gridDim.x * 32;
  for (int base = blockIdx.x * 32; base < E; base += stride) {
    const int e = base + lane;
    int et = -1;
    if (e < E) et = etype[e];
    const bool match = (et == r);
    const unsigned mask = __builtin_amdgcn_ballot_w32(match);
    if (match) {
      const int pos = qcount + __popc(mask & ((1u << lane) - 1u));
      qsrc[pos] = src[e];
      qdst[pos] = dst[e];
    }
    qcount += __popc(mask);
    __syncthreads();
    while (qcount >= 16) {
      rgcn_emit_tile(h1, h2acc, Ah, qsrc, qdst, bfrag, 16, lane);
      const int rem = qcount - 16;
      int ts = 0, td = 0;
      if (lane < rem) { ts = qsrc[16 + lane]; td = qdst[16 + lane]; }
      __syncthreads();
      if (lane < rem) { qsrc[lane] = ts; qdst[lane] = td; }
      qcount = rem;
      __syncthreads();
    }
  }
  if (qcount > 0)
    rgcn_emit_tile(h1, h2acc, Ah, qsrc, qdst, bfrag, qcount, lane);
}

// ---------------------------------------------------------------------------
// Kernel 4: fused layer-2 epilogue + layer-3 GEMM:
//   out = relu(h2acc + b2) @ W3 + b3, one 16-row tile per wave, f32 WMMA.
// ---------------------------------------------------------------------------
__global__ __launch_bounds__(32) void k_layer3_wmma(
    const float* __restrict__ h2acc, const float* __restrict__ b2,
    const float* __restrict__ W3, const float* __restrict__ b3,
    float* __restrict__ out, int N) {
  const int lane = threadIdx.x;
  const int l15 = lane & 15;
  const int half = lane >> 4;
  const int v0 = blockIdx.x * 16;

  v2f bfrag[8][2];
#pragma unroll
  for (int ks = 0; ks < 8; ++ks) {
    const int kb = ks * 4 + 2 * half;
#pragma unroll
    for (int nt = 0; nt < 2; ++nt) {
      const int n = nt * 16 + l15;
      bfrag[ks][nt].x = W3[(kb + 0) * DIM + n];
      bfrag[ks][nt].y = W3[(kb + 1) * DIM + n];
    }
  }

  const int rowA = v0 + l15;
  const bool okA = rowA < N;
  v8f c0 = {};
  v8f c1 = {};
#pragma unroll
  for (int ks = 0; ks < 8; ++ks) {
    const int kb = ks * 4 + 2 * half;
    v2f a;
    a.x = okA ? fmaxf(h2acc[(size_t)rowA * DIM + kb] + b2[kb], 0.0f) : 0.0f;
    a.y = okA ? fmaxf(h2acc[(size_t)rowA * DIM + kb + 1] + b2[kb + 1], 0.0f) : 0.0f;
    c0 = __builtin_amdgcn_wmma_f32_16x16x4_f32(false, a, false, bfrag[ks][0],
                                               (short)0, c0, false, false);
    c1 = __builtin_amdgcn_wmma_f32_16x16x4_f32(false, a, false, bfrag[ks][1],
                                               (short)0, c1, false, false);
  }

#pragma unroll
  for (int v = 0; v < 8; ++v) {
    const int row = v0 + v + 8 * half;
    if (row < N) {
      out[(size_t)row * DIM + l15] = c0[v] + b3[l15];
      out[(size_t)row * DIM + 16 + l15] = c1[v] + b3[16 + l15];
    }
  }
}

// ---------------------------------------------------------------------------
// Host launcher. Inputs (setup_inputs order):
//   0 src[E] i32, 1 dst[E] i32, 2 etype[E] i32, 3 n_nodes (device scalar,
//   unused: N derived from out_size), 4 W1[19*1*32], 5 b1[32],
//   6 W2[19*32*32], 7 b2[32], 8 W3[32*32], 9 b3[32]. Output: [N,32] f32.
// Workspace: cnt[19*N] | h2acc[N*32] | h1[N*32]  (~33.2 MB for N=100k).
// ---------------------------------------------------------------------------
extern "C" void kernel_launch(void* const* d_in, const int* in_sizes, int n_in,
                              void* d_out, int out_size, void* d_ws,
                              size_t ws_size, hipStream_t stream) {
  const int* src = (const int*)d_in[0];
  const int* dst = (const int*)d_in[1];
  const int* ety = (const int*)d_in[2];
  const float* W1 = (const float*)d_in[4];
  const float* b1 = (const float*)d_in[5];
  const float* W2 = (const float*)d_in[6];
  const float* b2 = (const float*)d_in[7];
  const float* W3 = (const float*)d_in[8];
  const float* b3 = (const float*)d_in[9];

  const int E = in_sizes[0];
  const int N = out_size / DIM;

  float* cnt = (float*)d_ws;                   // RELS*N
  float* h2a = cnt + (size_t)RELS * N;         // N*DIM accumulator
  float* h1 = h2a + (size_t)N * DIM;           // N*DIM layer-1 output

  const size_t zeroN = (size_t)RELS * N + (size_t)N * DIM;
  k_zero_f32<<<2048, 256, 0, stream>>>((float*)d_ws, zeroN);

  k_count_edges<<<(E + 255) / 256, 256, 0, stream>>>(dst, ety, cnt, E, N);

  const int l1_threads = N * DIM;
  k_layer1<<<(l1_threads + 255) / 256, 256, 0, stream>>>(cnt, W1, b1, h1, N);

  dim3 g2(1024, RELS);  // ~19k waves; ~10% tail-padding waste per block
  k_layer2_wmma<<<g2, 32, 0, stream>>>(src, dst, ety, h1, W2, h2a, E);

  k_layer3_wmma<<<(N + 15) / 16, 32, 0, stream>>>(h2a, b2, W3, b3,
                                                  (float*)d_out, N);
}